// Core_4544075399627
// MI455X (gfx1250) — compile-verified
//
#include <hip/hip_runtime.h>
#include <stdint.h>

#define NN 1024
#define EE (NN - 1)

typedef int v4i __attribute__((ext_vector_type(4)));
typedef int v8i __attribute__((ext_vector_type(8)));

__device__ __forceinline__ float sigmoidf(float x) {
    return 1.0f / (1.0f + __expf(-x));
}

// ---------------------------------------------------------------------------
// Kernel 1: vert = sigmoid(avg_border(radii)) * angles   -> (V,3) layout
// ---------------------------------------------------------------------------
__global__ __launch_bounds__(256) void k_vertices(const float* __restrict__ radii,
                                                  const float* __restrict__ angles,
                                                  float* __restrict__ vert) {
    const int n2 = NN * NN;
    int v = blockIdx.x * 256 + threadIdx.x;
    int f = v / n2;
    int rr = v - f * n2;
    int i = rr / NN;
    int j = rr - i * NN;

    // pairs: (fa,ka,la) <-> (fb,kb,lb); k: 0='r' (fixed row i), 1='c' (fixed col j)
    const int fa_[12] = {0, 0, 1, 1, 0, 0, 1, 1, 2, 2, 3, 3};
    const int ka_[12] = {0, 0, 0, 0, 1, 1, 1, 1, 1, 1, 1, 1};
    const int la_[12] = {EE, 0, EE, 0, EE, 0, EE, 0, EE, 0, EE, 0};
    const int fb_[12] = {2, 3, 2, 3, 4, 5, 4, 5, 4, 5, 4, 5};
    const int kb_[12] = {0, 0, 0, 0, 0, 0, 0, 0, 1, 1, 1, 1};
    const int lb_[12] = {EE, EE, 0, 0, EE, EE, 0, 0, EE, EE, 0, 0};

    int pf = -1, pi = 0, pj = 0;  // partner coords; last matching pair wins
#pragma unroll
    for (int p = 0; p < 12; ++p) {
        if (f == fa_[p] && ((ka_[p] == 0) ? (i == la_[p]) : (j == la_[p]))) {
            int pos = (ka_[p] == 0) ? j : i;
            pf = fb_[p];
            if (kb_[p] == 0) { pi = lb_[p]; pj = pos; } else { pi = pos; pj = lb_[p]; }
        }
        if (f == fb_[p] && ((kb_[p] == 0) ? (i == lb_[p]) : (j == lb_[p]))) {
            int pos = (kb_[p] == 0) ? j : i;
            pf = fa_[p];
            if (ka_[p] == 0) { pi = la_[p]; pj = pos; } else { pi = pos; pj = la_[p]; }
        }
    }

    float rad = radii[v];
    if (pf >= 0) rad = 0.5f * (rad + radii[pf * n2 + pi * NN + pj]);
    float s = sigmoidf(rad);

    int base = i * NN + j;
    vert[v * 3 + 0] = s * angles[(f * 3 + 0) * n2 + base];
    vert[v * 3 + 1] = s * angles[(f * 3 + 1) * n2 + base];
    vert[v * 3 + 2] = s * angles[(f * 3 + 2) * n2 + base];
}

// ---------------------------------------------------------------------------
// Kernel 2: faces indices -> float (exact for values < 2^24), 4-wide.
// ---------------------------------------------------------------------------
__global__ __launch_bounds__(256) void k_faces4(const int* __restrict__ faces,
                                                float* __restrict__ out, int n4) {
    int t = blockIdx.x * 256 + threadIdx.x;
    if (t < n4) {
        int4 a = ((const int4*)faces)[t];
        float4 r;
        r.x = (float)a.x; r.y = (float)a.y; r.z = (float)a.z; r.w = (float)a.w;
        ((float4*)out)[t] = r;
    }
}

__global__ void k_faces_tail(const int* __restrict__ faces, float* __restrict__ out,
                             int start, int n) {
    int t = start + blockIdx.x * 64 + threadIdx.x;
    if (t < n) out[t] = (float)faces[t];
}

// ---------------------------------------------------------------------------
// Kernel 3: vertex normals via analytic gather of <=6 incident triangles.
// Tile of vert (18x18 vertices = 18 rows x 54 dwords) staged into LDS by the
// Tensor Data Mover (tensor_load_to_lds), waited with s_wait_tensorcnt.
// ---------------------------------------------------------------------------
__global__ __launch_bounds__(256) void k_normals(const float* __restrict__ vert,
                                                 float* __restrict__ vn) {
    __shared__ __align__(16) float tile[18 * 54];
    const int n2 = NN * NN;

    int f  = blockIdx.y >> 6;          // 64 row-tiles per face
    int i0 = (blockIdx.y & 63) << 4;
    int j0 = blockIdx.x << 4;

    int rowBase = f * NN + i0 - 1; if (rowBase < 0) rowBase = 0;  // global row of halo
    int colBase = j0 - 1;          if (colBase < 0) colBase = 0;  // vertex col of halo

    if (threadIdx.x < 32) {  // wave 0 issues the TDM load
        uint64_t gaddr = (uint64_t)(uintptr_t)vert +
                         ((uint64_t)rowBase * (3 * NN) + (uint64_t)colBase * 3) * 4ull;
        unsigned ldsOff = (unsigned)(uintptr_t)(void*)tile;  // low 32b of flat = LDS offset

        v4i g0;
        g0[0] = 1;                                   // count=1 valid descriptor
        g0[1] = (int)ldsOff;                         // lds_addr (bytes)
        g0[2] = (int)(unsigned)(gaddr & 0xFFFFFFFFu);
        g0[3] = (int)(((unsigned)((gaddr >> 32) & 0x1FFFFFFu)) | (2u << 30)); // type=2

        const unsigned td0 = 3 * NN;                 // tensor dim0: 3072 dwords / row
        const unsigned td1 = 8 * NN;                 // tensor dim1 (never hit)
        v8i g1;
        g1[0] = (int)(2u << 16);                     // data_size = 4 bytes
        g1[1] = (int)((td0 & 0xFFFFu) << 16);        // tensor_dim0[15:0]
        g1[2] = (int)((td0 >> 16) | ((td1 & 0xFFFFu) << 16));
        g1[3] = (int)((td1 >> 16) | (54u << 16));    // tile_dim0 = 54 dwords
        g1[4] = 18;                                  // tile_dim1 = 18 rows, tile_dim2 = 0
        g1[5] = 3 * NN;                              // tensor_dim0_stride[31:0] = 3072
        g1[6] = 0;
        g1[7] = 0;
        v4i gz4 = {0, 0, 0, 0};
        v8i gz8 = {0, 0, 0, 0, 0, 0, 0, 0};
        // clang-23 / therock-10.0 form: 6 args (g0, g1, g2, g3, extra v8i, cpol)
        __builtin_amdgcn_tensor_load_to_lds(g0, g1, gz4, gz4, gz8, 0);
        __builtin_amdgcn_s_wait_tensorcnt(0);
    }
    __syncthreads();

    int tx = threadIdx.x & 15;
    int ty = threadIdx.x >> 4;
    int i = i0 + ty, j = j0 + tx;
    int gr = f * NN + i;

    auto ld = [&](int r_, int c_, float3& p) {
        int o = (r_ - rowBase) * 54 + (c_ - colBase) * 3;
        p.x = tile[o]; p.y = tile[o + 1]; p.z = tile[o + 2];
    };

    float ax = 0.f, ay = 0.f, az = 0.f;
    auto addTri = [&](const float3& a, const float3& b, const float3& c) {
        float ux = b.x - a.x, uy = b.y - a.y, uz = b.z - a.z;
        float vx = c.x - a.x, vy = c.y - a.y, vz = c.z - a.z;
        float cx = uy * vz - uz * vy;
        float cy = uz * vx - ux * vz;
        float cz = ux * vy - uy * vx;
        float l = sqrtf(cx * cx + cy * cy + cz * cz);
        float inv = 1.0f / fmaxf(l, 1e-8f);
        ax += cx * inv; ay += cy * inv; az += cz * inv;
    };

    bool iL = (i > 0), iH = (i < EE), jL = (j > 0), jH = (j < EE);
    float3 P00; ld(gr, j, P00);
    float3 Pjp{}, Ppj{}, Pmj{}, Pmjp{}, Pjm{}, Ppjm{};
    if (jH)       ld(gr,     j + 1, Pjp);
    if (iH)       ld(gr + 1, j,     Ppj);
    if (iL)       ld(gr - 1, j,     Pmj);
    if (iL && jH) ld(gr - 1, j + 1, Pmjp);
    if (jL)       ld(gr,     j - 1, Pjm);
    if (iH && jL) ld(gr + 1, j - 1, Ppjm);

    if (iH && jH)   addTri(P00, Pjp, Ppj);       // quad(i,j)     t1
    if (iL && jH) { addTri(Pmj, Pmjp, P00);      // quad(i-1,j)   t1
                    addTri(P00, Pmjp, Pjp); }    // quad(i-1,j)   t2
    if (iL && jL)   addTri(Pjm, Pmj, P00);       // quad(i-1,j-1) t2
    if (iH && jL) { addTri(Pjm, P00, Ppjm);      // quad(i,j-1)   t1
                    addTri(Ppjm, P00, Ppj); }    // quad(i,j-1)   t2

    float l = sqrtf(ax * ax + ay * ay + az * az);
    float inv = 1.0f / fmaxf(l, 1e-8f);
    float sgn = (f == 0 || f == 3 || f == 4) ? -1.0f : 1.0f;  // signs per cube face
    inv *= sgn;

    int v = f * n2 + i * NN + j;
    vn[v * 3 + 0] = ax * inv;
    vn[v * 3 + 1] = ay * inv;
    vn[v * 3 + 2] = az * inv;
}

// ---------------------------------------------------------------------------
extern "C" void kernel_launch(void* const* d_in, const int* in_sizes, int n_in,
                              void* d_out, int out_size, void* d_ws, size_t ws_size,
                              hipStream_t stream) {
    const float* radii  = (const float*)d_in[0];
    const float* angles = (const float*)d_in[1];
    const int*   faces  = (const int*)d_in[2];

    const int n2 = NN * NN;
    const int V  = 6 * n2;
    const int F3 = in_sizes[2];

    float* out    = (float*)d_out;
    float* vert   = out;                       // V*3
    float* ofaces = out + (size_t)V * 3;       // F3
    float* vn     = ofaces + (size_t)F3;       // V*3

    k_vertices<<<dim3((6 * n2) / 256), dim3(256), 0, stream>>>(radii, angles, vert);

    int n4 = F3 >> 2;                          // 128-bit vectorized body
    if (n4 > 0)
        k_faces4<<<dim3((n4 + 255) / 256), dim3(256), 0, stream>>>(faces, out + (size_t)V * 3, n4);
    if (F3 & 3)                                // scalar tail (F3 % 4 elements)
        k_faces_tail<<<dim3(1), dim3(64), 0, stream>>>(faces, out + (size_t)V * 3, n4 << 2, F3);

    k_normals<<<dim3(NN / 16, 6 * (NN / 16)), dim3(256), 0, stream>>>(vert, vn);
}